// PriorityLoss_17652315587318
// MI455X (gfx1250) — compile-verified
//
#include <hip/hip_runtime.h>

// PriorityLoss for MI455X (gfx1250, wave32).
// Pair-ranking stage uses V_WMMA_F32_16X16X4_F32 to generate the rank-2
// outer-difference tiles (p_i - p_j, t_i - t_j, raw_i + raw_j) on the matrix
// pipe; VALU does the hinge/mask math on the 8 accumulator regs per lane.

typedef float v2f __attribute__((ext_vector_type(2)));
typedef float v8f __attribute__((ext_vector_type(8)));

#define N_ELEM    8192
#define TILE      16
#define T_TILES   (N_ELEM / TILE)      // 512 tile rows/cols
#define K1_BLOCKS 32
#define K1_THREADS 256
#define K2_BLOCKS (T_TILES / 2)        // 256, triangle-balanced (rows b and 511-b)
#define K2_THREADS 256
#define K2_WAVES  (K2_THREADS / 32)

// Workspace layout (floats):
//   [0, 640)        : k1 per-block partials, 20 per block * 32 blocks
//   [640, 896)      : k2 per-block violation sums (float)
//   [896, 1152)     : k2 per-block pair counts (uint bits)
#define WS_K2V 640
#define WS_K2C 896

__device__ __forceinline__ float raw_w(float snr) {
    return 2.0f * __expf(-snr * (1.0f / 15.0f)) + 0.5f;
}

// ---------------- Kernel 1: elementwise statistics ----------------
__global__ void k1_stats(const float* __restrict__ p, const float* __restrict__ t,
                         const float* __restrict__ u, const float* __restrict__ s,
                         float* __restrict__ ws) {
    const int tid = blockIdx.x * blockDim.x + threadIdx.x;  // exactly N_ELEM threads
    float pv = p[tid], tv = t[tid], uv = u[tid], sv = s[tid];
    float raw = raw_w(sv);
    float sm  = tv * 0.95f + 0.025f;           // label smoothing
    float d   = pv - sm;
    float err = fabsf(pv - tv);
    float e   = fminf(fmaxf(err, 0.001f), 1.0f);
    float du  = uv - e;

    float vals[20];
    vals[0]  = raw;
    vals[1]  = raw * d * d;
    vals[2]  = uv;
    vals[3]  = uv * uv;
    vals[4]  = e;
    vals[5]  = e * e;
    vals[6]  = uv * e;
    vals[7]  = du * du;
    vals[8]  = fmaxf(0.01f - uv, 0.0f);
    vals[9]  = fmaxf(uv - 0.5f, 0.0f);
    vals[10] = pv;
    vals[11] = tv;
    vals[12] = pv * pv;
    vals[13] = tv * tv;
    vals[14] = fmaxf(-pv, 0.0f);
    vals[15] = fmaxf(pv - 1.0f, 0.0f);
    vals[16] = pv;  // max p
    vals[17] = tv;  // max t
    vals[18] = pv;  // min p
    vals[19] = tv;  // min t

    __shared__ float red[K1_THREADS];
    for (int q = 0; q < 20; ++q) {
        red[threadIdx.x] = vals[q];
        __syncthreads();
        for (int off = K1_THREADS / 2; off > 0; off >>= 1) {
            if (threadIdx.x < off) {
                float a = red[threadIdx.x], b = red[threadIdx.x + off];
                red[threadIdx.x] = (q < 16) ? (a + b)
                                 : (q < 18) ? fmaxf(a, b) : fminf(a, b);
            }
            __syncthreads();
        }
        if (threadIdx.x == 0) ws[blockIdx.x * 20 + q] = red[0];
        __syncthreads();
    }
}

// ---------------- Kernel 2: all-pairs ranking via WMMA tiles ----------------
__global__ void k2_rank(const float* __restrict__ p, const float* __restrict__ t,
                        const float* __restrict__ s,
                        float* __restrict__ wsV, unsigned* __restrict__ wsC) {
    const int lane = threadIdx.x & 31;
    const int wave = threadIdx.x >> 5;
    const int b    = blockIdx.x;

    const int rowA = b;                 // tile row b     : 512-b tiles
    const int rowB = T_TILES - 1 - b;   // tile row 511-b : b+1 tiles
    const int nA   = T_TILES - rowA;
    const int nTot = T_TILES + 1;       // 513 total tiles for this block

    const bool  lo  = lane < 16;
    const int   m   = lane & 15;        // A row (lanes 0-15) / B column index
    const float lof = lo ? 1.0f : 0.0f;

    // A operands (16x4 f32: VGPR0 = K0 lanes0-15; K2 lanes16-31 -> zeroed).
    // A[m] = (x_m, 1, 0, 0)
    const int iA0 = rowA * TILE, iB0 = rowB * TILE;
    float pA = p[iA0 + m], tA = t[iA0 + m], rA = raw_w(s[iA0 + m]);
    float pB = p[iB0 + m], tB = t[iB0 + m], rB = raw_w(s[iB0 + m]);
    pA *= lof; tA *= lof; rA *= lof;
    pB *= lof; tB *= lof; rB *= lof;
    const v2f aP_A = {pA, lof}, aT_A = {tA, lof}, aW_A = {rA, lof};
    const v2f aP_B = {pB, lof}, aT_B = {tB, lof}, aW_B = {rB, lof};
    const v8f zero = {0.f, 0.f, 0.f, 0.f, 0.f, 0.f, 0.f, 0.f};

    float    sumV = 0.0f;
    unsigned cnt  = 0u;

    for (int k = wave; k < nTot; k += K2_WAVES) {   // wave-uniform control flow
        int ti, tj;
        v2f aP, aT, aW;
        if (k < nA) { ti = rowA; tj = rowA + k;        aP = aP_A; aT = aT_A; aW = aW_A; }
        else        { ti = rowB; tj = rowB + (k - nA); aP = aP_B; aT = aT_B; aW = aW_B; }

        const int j0 = tj * TILE;
        float pj = p[j0 + m] * lof;
        float tjv = t[j0 + m] * lof;
        float rj = raw_w(s[j0 + m]) * lof;

        // B (4x16 f32): row K0 = ones, row K1 = (-p_n / -t_n / raw_n); K2/K3 = 0.
        const v2f bP = {lof, -pj};
        const v2f bT = {lof, -tjv};
        const v2f bW = {lof, rj};

        // D[m][n] = x_m - y_n (pd, td) and raw_m + raw_n (w), full f32.
        v8f pd = __builtin_amdgcn_wmma_f32_16x16x4_f32(false, aP, false, bP,
                                                       (short)0, zero, false, false);
        v8f td = __builtin_amdgcn_wmma_f32_16x16x4_f32(false, aT, false, bT,
                                                       (short)0, zero, false, false);
        v8f w  = __builtin_amdgcn_wmma_f32_16x16x4_f32(false, aW, false, bW,
                                                       (short)0, zero, false, false);

        const bool diag = (ti == tj);   // only diag tile needs per-element i<j
#pragma unroll
        for (int e = 0; e < 8; ++e) {
            const int   mIdx = e + (lo ? 0 : 8);
            const float tde  = td[e];
            const float pde  = pd[e];
            const float we   = w[e];
            const float atd  = fabsf(tde);
            const bool  msk  = (atd >= 0.05f) && (!diag || (mIdx < m));
            const float marg = 0.16f * fminf(fmaxf(atd, 0.1f), 1.0f);  // 2*0.08*clip
            const float spd  = (tde > 0.0f) ? pde : -pde;              // sign(td)*pd
            const float viol = fmaxf(marg - spd, 0.0f);
            sumV += msk ? we * viol : 0.0f;
            cnt  += msk ? 1u : 0u;
        }
    }

    __shared__ float    rv[K2_THREADS];
    __shared__ unsigned rc[K2_THREADS];
    rv[threadIdx.x] = sumV;
    rc[threadIdx.x] = cnt;
    __syncthreads();
    for (int off = K2_THREADS / 2; off > 0; off >>= 1) {
        if (threadIdx.x < off) {
            rv[threadIdx.x] += rv[threadIdx.x + off];
            rc[threadIdx.x] += rc[threadIdx.x + off];
        }
        __syncthreads();
    }
    if (threadIdx.x == 0) { wsV[b] = rv[0]; wsC[b] = rc[0]; }
}

// ---------------- Kernel 3: deterministic finalize ----------------
__global__ void k3_final(const float* __restrict__ ws, float* __restrict__ out) {
    if (threadIdx.x != 0 || blockIdx.x != 0) return;

    float S[16];
    for (int q = 0; q < 16; ++q) {
        float acc = 0.0f;
        for (int b2 = 0; b2 < K1_BLOCKS; ++b2) acc += ws[b2 * 20 + q];
        S[q] = acc;
    }
    float pmax = -1e30f, tmax = -1e30f, pmin = 1e30f, tmin = 1e30f;
    for (int b2 = 0; b2 < K1_BLOCKS; ++b2) {
        pmax = fmaxf(pmax, ws[b2 * 20 + 16]);
        tmax = fmaxf(tmax, ws[b2 * 20 + 17]);
        pmin = fminf(pmin, ws[b2 * 20 + 18]);
        tmin = fminf(tmin, ws[b2 * 20 + 19]);
    }
    float sumV = 0.0f;
    for (int b2 = 0; b2 < K2_BLOCKS; ++b2) sumV += ws[WS_K2V + b2];
    const unsigned* cptr = (const unsigned*)(ws + WS_K2C);
    unsigned cnt = 0u;
    for (int b2 = 0; b2 < K2_BLOCKS; ++b2) cnt += cptr[b2];

    const float invN = 1.0f / (float)N_ELEM;
    const float mean_raw = S[0] * invN;
    const float mrc = fmaxf(mean_raw, 1e-6f);

    // mse_loss = mean(snr_w * d^2), snr_w = raw/mean_raw
    const float mse_loss = (S[1] * invN) / mrc;

    // ranking: pair_w = 0.5*(raw_i+raw_j)/mean_raw; sumV holds sum (raw_i+raw_j)*viol
    const float total_rank = 0.5f * sumV / mrc;
    const float cntf = (float)cnt;
    const float ranking_loss = (cnt > 0u) ? total_rank / fmaxf(cntf, 1.0f) : 0.0f;

    // uncertainty
    const float mean_u = S[2] * invN, mean_e = S[4] * invN;
    const float mse_unc = S[7] * invN;
    const float cov = S[6] * invN - mean_u * mean_e;
    const float std_u = fmaxf(sqrtf(fmaxf(S[3] * invN - mean_u * mean_u, 0.0f)), 1e-6f);
    const float std_e = fmaxf(sqrtf(fmaxf(S[5] * invN - mean_e * mean_e, 0.0f)), 1e-6f);
    const float corr = cov / (std_u * std_e + 1e-6f);
    const float cl = fmaxf(0.5f - corr, 0.0f);
    const float unc_loss = 0.5f * mse_unc + 0.3f * cl * cl
                         + 0.2f * (fmaxf(-corr, 0.0f) * 10.0f);

    const float unc_bounds = 0.05f * (S[8] * invN + S[9] * invN);

    // calibration
    const float mean_p = S[10] * invN, mean_t = S[11] * invN;
    const float mean_gap = fabsf(mean_p - mean_t);
    const float max_gap = fmaxf(1.0f - (pmax + 1e-6f) / (tmax + 1e-6f), 0.0f);
    const float p_range = pmax - pmin, t_range = tmax - tmin;
    const float range_pen = fmaxf(1.0f - (p_range + 1e-6f) / (t_range + 1e-6f), 0.0f);
    const float calib = 0.2f * mean_gap + 1.5f * max_gap + 1.0f * range_pen;

    // variance penalty
    const float p_std = sqrtf(fmaxf(S[12] * invN - mean_p * mean_p, 0.0f));
    const float t_std = sqrtf(fmaxf(S[13] * invN - mean_t * mean_t, 0.0f));
    const float vr = p_std / (t_std + 1e-8f);
    const float mvp = ((p_std < 0.5f * t_std) && (t_std > 1e-4f))
                    ? 2.0f * fmaxf(0.5f - vr, 0.0f) : 0.0f;

    const float bpen = 5.0f * (S[14] * invN + S[15] * invN);

    // bucket5 == true: eff weights 0.5*0.85 and 0.4*1.5
    const float total = 0.425f * mse_loss + 0.6f * ranking_loss
                      + 0.35f * unc_loss + unc_bounds + calib + mvp + bpen;
    out[0] = total;
}

extern "C" void kernel_launch(void* const* d_in, const int* in_sizes, int n_in,
                              void* d_out, int out_size, void* d_ws, size_t ws_size,
                              hipStream_t stream) {
    (void)in_sizes; (void)n_in; (void)out_size; (void)ws_size;
    const float* p = (const float*)d_in[0];
    const float* t = (const float*)d_in[1];
    const float* u = (const float*)d_in[2];
    const float* s = (const float*)d_in[3];
    float* ws = (float*)d_ws;

    k1_stats<<<K1_BLOCKS, K1_THREADS, 0, stream>>>(p, t, u, s, ws);
    k2_rank<<<K2_BLOCKS, K2_THREADS, 0, stream>>>(p, t, s,
                                                  ws + WS_K2V,
                                                  (unsigned*)(ws + WS_K2C));
    k3_final<<<1, 1, 0, stream>>>(ws, (float*)d_out);
}